// BirthDeathLoss_40295383171311
// MI455X (gfx1250) — compile-verified
//
#include <hip/hip_runtime.h>

// ---------------------------------------------------------------------------
// BirthDeathLoss for MI455X (gfx1250, wave32)
//
// loss = ALPHA*L0 + (1-ALPHA)*L1
// L = (1/(B*C)) * sum_{b,c,k} [ good ? BETA*(1-diff) : (1-BETA)*diff ]
// diff = (pred[birth] - pred[death])^2,  good = (k < betti[b,c])
//
// Memory/latency-bound gather+reduce (~130 MB HBM traffic -> ~5.6 us floor
// at 23.3 TB/s; ~10 MFLOP total, so no matmul structure). CDNA5 paths used:
//  - v_wmma_f32_16x16x4_f32 (all-ones B) as the exact wave32 f32 reduction
//  - global_prefetch_b8 on the streaming index data
//  - nontemporal (NT) loads for the single-use index stream; default RT
//    caching for the gathers (prediction fits in the 192 MB L2)
// ---------------------------------------------------------------------------

typedef float v2f __attribute__((ext_vector_type(2)));
typedef float v8f __attribute__((ext_vector_type(8)));
typedef int   v4i __attribute__((ext_vector_type(4)));

#define BB     32
#define HH     1024
#define WW     1024
#define KK     16384
#define ALPHA  0.5f
#define BETA   0.5f

#define NBLOCKS  1024
#define NTHREADS 256
#define NTHR     (NBLOCKS * NTHREADS)          // 262144 = 2^18
#define ITERS_PER_COMP ((BB * KK) / NTHR)      // 524288 / 262144 = 2

// Full-precision wave32 sum using the f32 WMMA.
// A-matrix 16x4 layout: lanes 0-15 -> {v0:K=0, v1:K=1}, lanes 16-31 -> {v0:K=2, v1:K=3}.
// With B = ones, D[m][n] = sum_k A[m][k] = partial[m] + partial[m+16].
// C/D layout: lane n<16 holds D[0..7][n] in its 8 VGPRs, lane n+16 holds D[8..15][n].
// So (my 8-element sum) + (same from lane^16) == sum over all 32 lanes.
// Requires EXEC all-ones: call only from fully convergent code.
__device__ __forceinline__ float wave_reduce_wmma(float x) {
  v2f a; a[0] = x;    a[1] = 0.0f;
  v2f b; b[0] = 1.0f; b[1] = 1.0f;
  v8f c = {};
  c = __builtin_amdgcn_wmma_f32_16x16x4_f32(
      /*neg_a=*/false, a, /*neg_b=*/false, b,
      /*c_mod=*/(short)0, c, /*reuse_a=*/false, /*reuse_b=*/false);
  float s = ((c[0] + c[1]) + (c[2] + c[3])) + ((c[4] + c[5]) + (c[6] + c[7]));
  s += __shfl_xor(s, 16, 32);
  return s;
}

__global__ __launch_bounds__(NTHREADS) void bdloss_main(
    const float* __restrict__ pred,
    const int*  __restrict__ iv0,   // (B,C,K,2,2) int32, 4 coords per interval
    const int*  __restrict__ iv1,
    const int*  __restrict__ bt0,   // (B,C) int32
    const int*  __restrict__ bt1,
    float* __restrict__ partials)   // one slot per block (d_ws)
{
  const int tid = blockIdx.x * NTHREADS + threadIdx.x;   // < 2^18

  float acc = 0.0f;

  // Explicit component loop: iv/bt/weights are loop-invariant scalars
  // (SGPR-resident), no per-interval cndmask selects. Every thread runs the
  // identical trip count -> EXEC stays all-ones for the WMMA reduction.
  #pragma unroll
  for (int comp = 0; comp < 2; ++comp) {
    const int*  iv = comp ? iv1 : iv0;
    const int*  bt = comp ? bt1 : bt0;
    const float f  = (comp ? (1.0f - ALPHA) : ALPHA) / (float)BB;  // /(B*C)
    const float wg = f * BETA;            // coefficient for "good" intervals
    const float wb = f * (1.0f - BETA);   // coefficient for "bad" intervals

    #pragma unroll
    for (int it = 0; it < ITERS_PER_COMP; ++it) {
      const int rem  = tid + it * NTHR;        // interval index in component
      const int bimg = rem >> 14;              // / K (K = 2^14)
      const int k    = rem & (KK - 1);

      const v4i* vp = (const v4i*)iv + rem;

      // Speculative prefetch of the index stream one chunk ahead (gfx1250
      // global_prefetch_b8; OOB speculative prefetches are silently dropped).
      __builtin_prefetch((const void*)(vp + NTHR), 0, 0);

      // Index stream is single-use per call: non-temporal so it does not
      // evict gather lines from L2.
      const v4i v = __builtin_nontemporal_load(vp); // [r_b, c_b, r_d, c_d]

      // Gathers: default (RT) caching -- prediction plane lives in L2.
      const float* p = pred + (long long)bimg * (HH * WW);
      const float birth = p[v[0] * WW + v[1]];
      const float death = p[v[2] * WW + v[3]];
      const float d     = birth - death;
      const float diff  = d * d;

      const bool good = k < bt[bimg];
      // good: wg*(1-diff) = wg - wg*diff ; bad: wb*diff
      acc += good ? fmaf(-wg, diff, wg) : wb * diff;
    }
  }

  // wave32 reduce via WMMA, then cross-wave reduce via LDS (deterministic).
  const float wsum = wave_reduce_wmma(acc);

  __shared__ float lds[NTHREADS / 32];
  const int wave = threadIdx.x >> 5;
  const int lane = threadIdx.x & 31;
  if (lane == 0) lds[wave] = wsum;
  __syncthreads();

  if (threadIdx.x == 0) {
    float t = 0.0f;
    #pragma unroll
    for (int i = 0; i < NTHREADS / 32; ++i) t += lds[i];
    partials[blockIdx.x] = t;   // every slot overwritten each call
  }
}

__global__ __launch_bounds__(NTHREADS) void bdloss_final(
    const float* __restrict__ partials,
    float* __restrict__ out)
{
  float acc = 0.0f;
  #pragma unroll
  for (int it = 0; it < NBLOCKS / NTHREADS; ++it)   // 4 uniform iterations
    acc += partials[threadIdx.x + it * NTHREADS];

  const float wsum = wave_reduce_wmma(acc);

  __shared__ float lds[NTHREADS / 32];
  const int wave = threadIdx.x >> 5;
  const int lane = threadIdx.x & 31;
  if (lane == 0) lds[wave] = wsum;
  __syncthreads();

  if (threadIdx.x == 0) {
    float t = 0.0f;
    #pragma unroll
    for (int i = 0; i < NTHREADS / 32; ++i) t += lds[i];
    out[0] = t;                 // overwrite: no dependence on poisoned state
  }
}

extern "C" void kernel_launch(void* const* d_in, const int* in_sizes, int n_in,
                              void* d_out, int out_size, void* d_ws, size_t ws_size,
                              hipStream_t stream) {
  const float* pred = (const float*)d_in[0];
  const int*   iv0  = (const int*)d_in[1];
  const int*   iv1  = (const int*)d_in[2];
  const int*   bt0  = (const int*)d_in[3];
  const int*   bt1  = (const int*)d_in[4];
  float* out        = (float*)d_out;
  float* partials   = (float*)d_ws;           // NBLOCKS floats = 4 KB

  bdloss_main<<<NBLOCKS, NTHREADS, 0, stream>>>(pred, iv0, iv1, bt0, bt1, partials);
  bdloss_final<<<1, NTHREADS, 0, stream>>>(partials, out);
}